// MultiHeadedAttention_44238163149251
// MI455X (gfx1250) — compile-verified
//
#include <hip/hip_runtime.h>
#include <hip/hip_bf16.h>
#include <math.h>

typedef __attribute__((ext_vector_type(2))) float v2f;
typedef __attribute__((ext_vector_type(8))) float v8f;

#define N_SEQ 2048
#define DM    512
#define HEADS 8
#define DHEAD 64

static __device__ __forceinline__ v8f wmma_f32_k4(v2f a, v2f b, v8f c) {
  // V_WMMA_F32_16X16X4_F32 : D = A(16x4) * B(4x16) + C(16x16), exact fp32
  return __builtin_amdgcn_wmma_f32_16x16x4_f32(false, a, false, b, (short)0, c, false, false);
}

// ---------------------------------------------------------------------------
// out[perm(o)][n] = sum_c W[o][c] * X[c][n] + bias[o]
// Each wave computes a 16(o) x 64(n) strip: 4 accumulators share one A frag.
// grid (N/256, M/16), block 128 (4 waves)
// ---------------------------------------------------------------------------
__global__ __launch_bounds__(128) void gemm512_kernel(
    const float* __restrict__ W, const float* __restrict__ bias,
    const float* __restrict__ X, float* __restrict__ out, int permute)
{
  const int tid    = threadIdx.x;
  const int wave   = tid >> 5;
  const int lane   = tid & 31;
  const int l      = lane & 15;
  const int hf     = lane >> 4;
  const int o0     = blockIdx.y * 16;
  const int n_base = blockIdx.x * 256 + wave * 64;

  v8f acc0 = {}, acc1 = {}, acc2 = {}, acc3 = {};
#pragma unroll 4
  for (int c0 = 0; c0 < DM; c0 += 4) {
    const int ca = c0 + 2 * hf;
    v2f a;
    a[0] = W[(o0 + l) * DM + ca];            // A[M=o][K=c], shared by 4 tiles
    a[1] = W[(o0 + l) * DM + ca + 1];
    // prefetch next K-slab of the streaming operand (global_prefetch_b8)
    if ((c0 & 15) == 0 && c0 + 16 < DM)
      __builtin_prefetch(&X[(c0 + 16 + 2 * hf) * N_SEQ + n_base + lane], 0, 1);
    v2f b0, b1, b2, b3;
    b0[0] = X[ca * N_SEQ + n_base + l];
    b0[1] = X[(ca + 1) * N_SEQ + n_base + l];
    b1[0] = X[ca * N_SEQ + n_base + 16 + l];
    b1[1] = X[(ca + 1) * N_SEQ + n_base + 16 + l];
    b2[0] = X[ca * N_SEQ + n_base + 32 + l];
    b2[1] = X[(ca + 1) * N_SEQ + n_base + 32 + l];
    b3[0] = X[ca * N_SEQ + n_base + 48 + l];
    b3[1] = X[(ca + 1) * N_SEQ + n_base + 48 + l];
    acc0 = wmma_f32_k4(a, b0, acc0);
    acc1 = wmma_f32_k4(a, b1, acc1);
    acc2 = wmma_f32_k4(a, b2, acc2);
    acc3 = wmma_f32_k4(a, b3, acc3);
  }
#pragma unroll
  for (int r = 0; r < 8; ++r) {
    const int o  = o0 + r + 8 * hf;
    const int po = permute ? ((o & 7) * DHEAD + (o >> 3)) : o;
    const float bo = bias[o];
    out[po * N_SEQ + n_base + l]      = acc0[r] + bo;
    out[po * N_SEQ + n_base + 16 + l] = acc1[r] + bo;
    out[po * N_SEQ + n_base + 32 + l] = acc2[r] + bo;
    out[po * N_SEQ + n_base + 48 + l] = acc3[r] + bo;
  }
}

// ---------------------------------------------------------------------------
// Pass 1: scores[h][n][m] = (Q_h^T K_h)/8 computed on the fly.
//   - mean_out[n][m] = (1/H) * sum_h scores   (each (n,m) owned by 1 thread)
//   - global min via ordered-uint atomicMin
// Each wave: 16(n) x 64(m) strip, 4 accumulators share one A (Q) fragment.
// grid (N/256, N/16), block 128
// ---------------------------------------------------------------------------
__global__ __launch_bounds__(128) void score_stats_kernel(
    const float* __restrict__ qh, const float* __restrict__ kh,
    float* __restrict__ mean_out, unsigned int* __restrict__ min_key)
{
  __shared__ float wmin[4];
  const int tid    = threadIdx.x;
  const int wave   = tid >> 5;
  const int lane   = tid & 31;
  const int l      = lane & 15;
  const int hf     = lane >> 4;
  const int n0     = blockIdx.y * 16;
  const int m_base = blockIdx.x * 256 + wave * 64;

  v8f   msum0 = {}, msum1 = {}, msum2 = {}, msum3 = {};
  float lmin = INFINITY;
  for (int h = 0; h < HEADS; ++h) {
    const float* Q = qh + h * (DHEAD * N_SEQ);
    const float* K = kh + h * (DHEAD * N_SEQ);
    v8f acc0 = {}, acc1 = {}, acc2 = {}, acc3 = {};
#pragma unroll
    for (int d0 = 0; d0 < DHEAD; d0 += 4) {
      const int da = d0 + 2 * hf;
      v2f a;
      a[0] = Q[da * N_SEQ + n0 + l];          // A[M=n][K=d], shared
      a[1] = Q[(da + 1) * N_SEQ + n0 + l];
      v2f b0, b1, b2, b3;
      b0[0] = K[da * N_SEQ + m_base + l];
      b0[1] = K[(da + 1) * N_SEQ + m_base + l];
      b1[0] = K[da * N_SEQ + m_base + 16 + l];
      b1[1] = K[(da + 1) * N_SEQ + m_base + 16 + l];
      b2[0] = K[da * N_SEQ + m_base + 32 + l];
      b2[1] = K[(da + 1) * N_SEQ + m_base + 32 + l];
      b3[0] = K[da * N_SEQ + m_base + 48 + l];
      b3[1] = K[(da + 1) * N_SEQ + m_base + 48 + l];
      acc0 = wmma_f32_k4(a, b0, acc0);
      acc1 = wmma_f32_k4(a, b1, acc1);
      acc2 = wmma_f32_k4(a, b2, acc2);
      acc3 = wmma_f32_k4(a, b3, acc3);
    }
#pragma unroll
    for (int r = 0; r < 8; ++r) {
      const float s0 = acc0[r] * 0.125f;      // 1/sqrt(64)
      const float s1 = acc1[r] * 0.125f;
      const float s2 = acc2[r] * 0.125f;
      const float s3 = acc3[r] * 0.125f;
      msum0[r] += s0;  msum1[r] += s1;  msum2[r] += s2;  msum3[r] += s3;
      lmin = fminf(lmin, fminf(fminf(s0, s1), fminf(s2, s3)));
    }
  }
#pragma unroll
  for (int r = 0; r < 8; ++r) {
    const int n = n0 + r + 8 * hf;
    mean_out[n * N_SEQ + m_base + l]      = msum0[r] * 0.125f;   // /H
    mean_out[n * N_SEQ + m_base + 16 + l] = msum1[r] * 0.125f;
    mean_out[n * N_SEQ + m_base + 32 + l] = msum2[r] * 0.125f;
    mean_out[n * N_SEQ + m_base + 48 + l] = msum3[r] * 0.125f;
  }
  for (int off = 16; off > 0; off >>= 1)
    lmin = fminf(lmin, __shfl_xor(lmin, off, 32));
  if (lane == 0) wmin[wave] = lmin;
  __syncthreads();
  if (tid == 0) {
    const float v = fminf(fminf(wmin[0], wmin[1]), fminf(wmin[2], wmin[3]));
    const unsigned u   = __float_as_uint(v);
    const unsigned key = (u & 0x80000000u) ? ~u : (u | 0x80000000u);
    atomicMin(min_key, key);
  }
}

__global__ void init_kernel(unsigned int* min_key) {
  if (threadIdx.x == 0) *min_key = 0xFFFFFFFFu;
}

__global__ void finalize_min_kernel(const unsigned int* min_key, float* smin_f) {
  if (threadIdx.x == 0) {
    const unsigned key = *min_key;
    const unsigned u   = (key & 0x80000000u) ? (key ^ 0x80000000u) : ~key;
    *smin_f = __uint_as_float(u);
  }
}

// mean += smin * (~mask)  (mask independent of head => added once to the mean)
__global__ void fixup_mean_kernel(float* __restrict__ mean_out,
                                  const unsigned char* __restrict__ mask,
                                  const float* __restrict__ smin_f) {
  const int i = blockIdx.x * 256 + threadIdx.x;
  if (i < N_SEQ * N_SEQ) {
    if (!mask[i]) mean_out[i] += smin_f[0];
  }
}

// ---------------------------------------------------------------------------
// Pass 2: per (head, 16-query block): recompute masked scores into a 128 KB
// LDS strip (CDNA5 320KB WGP LDS), exact softmax, then x = P * V via WMMA.
// Q fragments for the whole K-sweep are hoisted into 32 VGPRs.
// Output written in merged layout xm[c = d*8 + h][n].
// grid (N/16, H), block 128
// ---------------------------------------------------------------------------
__global__ __launch_bounds__(128) void attn_kernel(
    const float* __restrict__ qh, const float* __restrict__ kh,
    const float* __restrict__ vh, const unsigned char* __restrict__ mask,
    const float* __restrict__ smin_p, float* __restrict__ xm)
{
  __shared__ float S[16][N_SEQ];   // 128 KB score/prob strip
  __shared__ float red[16][8];

  const int tid  = threadIdx.x;
  const int wave = tid >> 5;
  const int lane = tid & 31;
  const int l    = lane & 15;
  const int hf   = lane >> 4;
  const int h    = blockIdx.y;
  const int n0   = blockIdx.x * 16;
  const float smin = smin_p[0];

  const float* Q = qh + h * (DHEAD * N_SEQ);
  const float* K = kh + h * (DHEAD * N_SEQ);
  const float* V = vh + h * (DHEAD * N_SEQ);

  // ---- Phase A: masked scores into LDS; Q fragments loop-invariant ----
  v2f qa[16];
#pragma unroll
  for (int i = 0; i < 16; ++i) {
    const int da = 4 * i + 2 * hf;
    qa[i][0] = Q[da * N_SEQ + n0 + l];        // A[M=n][K=d]
    qa[i][1] = Q[(da + 1) * N_SEQ + n0 + l];
  }
  for (int m0 = wave * 16; m0 < N_SEQ; m0 += 64) {
    v8f acc = {};
#pragma unroll
    for (int i = 0; i < 16; ++i) {
      const int da = 4 * i + 2 * hf;
      v2f b;
      b[0] = K[da * N_SEQ + m0 + l];          // B[K=d][N=m]
      b[1] = K[(da + 1) * N_SEQ + m0 + l];
      acc = wmma_f32_k4(qa[i], b, acc);
    }
#pragma unroll
    for (int r = 0; r < 8; ++r) {
      const int n = r + 8 * hf;
      const int m = m0 + l;
      float s = acc[r] * 0.125f;
      s += mask[(n0 + n) * N_SEQ + m] ? 0.0f : smin;   // branchless cndmask
      S[n][m] = s;
    }
  }
  __syncthreads();

  // ---- Phase B: exact softmax per row (8 threads/row) ----
  {
    const int row = tid >> 3;
    const int t8  = tid & 7;
    float mx = -INFINITY;
    for (int m = t8; m < N_SEQ; m += 8) mx = fmaxf(mx, S[row][m]);
    red[row][t8] = mx;
    __syncthreads();
    if (t8 == 0) {
      float v = red[row][0];
#pragma unroll
      for (int i = 1; i < 8; ++i) v = fmaxf(v, red[row][i]);
      red[row][0] = v;
    }
    __syncthreads();
    mx = red[row][0];
    float sum = 0.0f;
    for (int m = t8; m < N_SEQ; m += 8) {
      const float e = __expf(S[row][m] - mx);
      S[row][m] = e;
      sum += e;
    }
    __syncthreads();
    red[row][t8] = sum;
    __syncthreads();
    if (t8 == 0) {
      float v = 0.0f;
#pragma unroll
      for (int i = 0; i < 8; ++i) v += red[row][i];
      red[row][0] = v;
    }
    __syncthreads();
    const float inv = 1.0f / red[row][0];
    for (int m = t8; m < N_SEQ; m += 8) S[row][m] *= inv;
  }
  __syncthreads();

  // ---- Phase C: x[n][d] = sum_m P[n][m] V[d][m], wave owns 16 dims ----
  {
    const int d0 = wave * 16;
    v8f acc = {};
    for (int m0 = 0; m0 < N_SEQ; m0 += 4) {
      const int ma = m0 + 2 * hf;
      v2f a, b;
      a[0] = S[l][ma];                       // A[M=n][K=m] from LDS
      a[1] = S[l][ma + 1];
      b[0] = V[(d0 + l) * N_SEQ + ma];       // B[K=m][N=d] = V[d][m]
      b[1] = V[(d0 + l) * N_SEQ + ma + 1];
      acc = wmma_f32_k4(a, b, acc);
    }
#pragma unroll
    for (int r = 0; r < 8; ++r) {
      const int n = n0 + r + 8 * hf;
      const int d = d0 + l;
      xm[(d * HEADS + h) * N_SEQ + n] = acc[r];   // merged channel c = d*8+h
    }
  }
}

// ---------------------------------------------------------------------------
extern "C" void kernel_launch(void* const* d_in, const int* in_sizes, int n_in,
                              void* d_out, int out_size, void* d_ws, size_t ws_size,
                              hipStream_t stream) {
  const float* query = (const float*)d_in[0];
  const float* key   = (const float*)d_in[1];
  const float* value = (const float*)d_in[2];
  /* d_in[3] = dist (unused by reference) */
  const unsigned char* mask = (const unsigned char*)d_in[4];
  const float* Wq = (const float*)d_in[5];
  const float* bq = (const float*)d_in[6];
  const float* Wk = (const float*)d_in[7];
  const float* bk = (const float*)d_in[8];
  const float* Wv = (const float*)d_in[9];
  const float* bv = (const float*)d_in[10];
  const float* Wm = (const float*)d_in[11];
  const float* bm = (const float*)d_in[12];

  float* out1 = (float*)d_out;             // [512, 2048] merged output
  float* out2 = out1 + DM * N_SEQ;         // [2048, 2048] scores.mean(axis=1)

  float* qh = (float*)d_ws;                         // [8][64][2048]
  float* kh = qh + HEADS * DHEAD * N_SEQ;
  float* vh = kh + HEADS * DHEAD * N_SEQ;
  float* xm = vh + HEADS * DHEAD * N_SEQ;           // [512][2048]
  unsigned int* min_key = (unsigned int*)(xm + DM * N_SEQ);
  float* smin_f = (float*)(min_key + 1);

  hipLaunchKernelGGL(init_kernel, dim3(1), dim3(32), 0, stream, min_key);

  const dim3 gGemm(N_SEQ / 256, DM / 16);   // (8, 32)
  hipLaunchKernelGGL(gemm512_kernel, gGemm, dim3(128), 0, stream, Wq, bq, query, qh, 1);
  hipLaunchKernelGGL(gemm512_kernel, gGemm, dim3(128), 0, stream, Wk, bk, key,   kh, 1);
  hipLaunchKernelGGL(gemm512_kernel, gGemm, dim3(128), 0, stream, Wv, bv, value, vh, 1);

  const dim3 gScore(N_SEQ / 256, N_SEQ / 16);  // (8, 128)
  hipLaunchKernelGGL(score_stats_kernel, gScore, dim3(128), 0, stream, qh, kh, out2, min_key);

  hipLaunchKernelGGL(finalize_min_kernel, dim3(1), dim3(32), 0, stream, min_key, smin_f);
  hipLaunchKernelGGL(fixup_mean_kernel, dim3((N_SEQ * N_SEQ) / 256), dim3(256), 0, stream,
                     out2, mask, smin_f);

  const dim3 gAttn(N_SEQ / 16, HEADS);        // (128, 8)
  hipLaunchKernelGGL(attn_kernel, gAttn, dim3(128), 0, stream, qh, kh, vh, mask, smin_f, xm);

  hipLaunchKernelGGL(gemm512_kernel, gGemm, dim3(128), 0, stream, Wm, bm, xm, out1, 0);
}